// SCLHead_52690658787380
// MI455X (gfx1250) — compile-verified
//
#include <hip/hip_runtime.h>
#include <math.h>

typedef __attribute__((ext_vector_type(2))) float v2f;
typedef __attribute__((ext_vector_type(8))) float v8f;

static constexpr int   D      = 128;       // feature dim
static constexpr int   KSTEPS = D / 4;     // 32 WMMA k-steps (K=4 each)
static constexpr int   NWAVES = 4;
static constexpr float INV_T  = 10.0f;     // 1 / temperature

// Merge two (max, sumexp) streaming-LSE partials.
__device__ __forceinline__ void lse_merge(float& m, float& s, float m2, float s2) {
    float mn = fmaxf(m, m2);
    if (mn == -__builtin_inff()) return;   // both partials empty
    s = s * __expf(m - mn) + s2 * __expf(m2 - mn);
    m = mn;
}

__global__ void zero_out_kernel(float* out) {
    if (threadIdx.x == 0 && blockIdx.x == 0) out[0] = 0.0f;
}

// One block = one 16-row anchor block. Two fused GEMM sweeps over all column
// tiles: pass 1 builds per-row negative logsumexp, pass 2 accumulates the
// positive-pair CE terms. f32 WMMA (16x16x4) keeps exact fp32 math.
__global__ __launch_bounds__(NWAVES * 32) void scl_loss_kernel(
    const float* __restrict__ f,      // [M, D]
    const int*   __restrict__ label,  // [M/2]
    float*       __restrict__ out,    // scalar
    int Mtot)
{
    const int tid  = threadIdx.x;
    const int wave = tid >> 5;
    const int lane = tid & 31;
    const int half = lane >> 4;   // 0: rows 0-7 of tile, 1: rows 8-15
    const int lmod = lane & 15;   // column-within-tile this lane owns
    const int r0   = blockIdx.x * 16;
    const int numTiles = Mtot >> 4;

    __shared__ float s_m[NWAVES][16];
    __shared__ float s_s[NWAVES][16];
    __shared__ float s_lse[16];
    __shared__ float s_l[NWAVES][16];
    __shared__ int   s_c[NWAVES][16];

    // ---- Preload A fragments: rows r0..r0+15, full K=128 (64 VGPRs). ----
    // f32 WMMA A layout: lane (16*h + m) vgpr j holds A(m, 2h + j).
    const float* arow = f + (size_t)(r0 + lmod) * D + 2 * half;
    v2f afrag[KSTEPS];
#pragma unroll
    for (int kk = 0; kk < KSTEPS; ++kk)
        afrag[kk] = *(const v2f*)(arow + 4 * kk);

    // Labels of the 8 anchor rows this lane's accumulators cover.
    int rlab[8];
#pragma unroll
    for (int i = 0; i < 8; ++i)
        rlab[i] = label[(r0 + 8 * half + i) >> 1];

    // =================== Pass 1: negative logsumexp ===================
    float rmax[8], rsum[8];
#pragma unroll
    for (int i = 0; i < 8; ++i) { rmax[i] = -__builtin_inff(); rsum[i] = 0.0f; }

    for (int t = wave; t < numTiles; t += NWAVES) {
        const int c0 = t * 16;
        const float* brow = f + (size_t)(c0 + lmod) * D + 2 * half;
        const int clab = label[(c0 + lmod) >> 1];

        v8f acc = {};
#pragma unroll
        for (int kk = 0; kk < KSTEPS; ++kk) {
            v2f b = *(const v2f*)(brow + 4 * kk);
            acc = __builtin_amdgcn_wmma_f32_16x16x4_f32(
                false, afrag[kk], false, b, (short)0, acc, false, false);
        }
        // Element i of acc = G(r0 + 8*half + i, c0 + lmod); stream into LSE.
#pragma unroll
        for (int i = 0; i < 8; ++i) {
            float x = acc[i] * INV_T;
            if (rlab[i] != clab) {                 // negative (diag never neg)
                float mn = fmaxf(rmax[i], x);
                rsum[i]  = rsum[i] * __expf(rmax[i] - mn) + __expf(x - mn);
                rmax[i]  = mn;
            }
        }
    }

    // XOR-shuffle reduce over the 16 lanes sharing each row set (wave32).
#pragma unroll
    for (int off = 1; off < 16; off <<= 1) {
#pragma unroll
        for (int i = 0; i < 8; ++i) {
            float m2 = __shfl_xor(rmax[i], off);
            float s2 = __shfl_xor(rsum[i], off);
            lse_merge(rmax[i], rsum[i], m2, s2);
        }
    }
    if (lmod == 0) {
#pragma unroll
        for (int i = 0; i < 8; ++i) {
            s_m[wave][8 * half + i] = rmax[i];
            s_s[wave][8 * half + i] = rsum[i];
        }
    }
    __syncthreads();
    if (wave == 0 && lane < 16) {
        float m = s_m[0][lane], s = s_s[0][lane];
#pragma unroll
        for (int w = 1; w < NWAVES; ++w) lse_merge(m, s, s_m[w][lane], s_s[w][lane]);
        s_lse[lane] = m + __logf(s);
    }
    __syncthreads();

    float nlse[8];
#pragma unroll
    for (int i = 0; i < 8; ++i) nlse[i] = s_lse[8 * half + i];

    // ============ Pass 2: positive-pair CE terms (recompute) ============
    float lsum[8];
    int   pcnt[8];
#pragma unroll
    for (int i = 0; i < 8; ++i) { lsum[i] = 0.0f; pcnt[i] = 0; }

    for (int t = wave; t < numTiles; t += NWAVES) {
        const int c0 = t * 16;
        const float* brow = f + (size_t)(c0 + lmod) * D + 2 * half;
        const int clab = label[(c0 + lmod) >> 1];
        const int gcol = c0 + lmod;

        v8f acc = {};
#pragma unroll
        for (int kk = 0; kk < KSTEPS; ++kk) {
            v2f b = *(const v2f*)(brow + 4 * kk);
            acc = __builtin_amdgcn_wmma_f32_16x16x4_f32(
                false, afrag[kk], false, b, (short)0, acc, false, false);
        }
#pragma unroll
        for (int i = 0; i < 8; ++i) {
            float x = acc[i] * INV_T;
            const int grow = r0 + 8 * half + i;
            if (rlab[i] == clab && grow != gcol) {      // positive, excl. self
                // logaddexp(x, L) - x, numerically stable both directions
                float d = nlse[i] - x;
                float term = (d > 0.0f) ? d + log1pf(__expf(-d))
                                        : log1pf(__expf(d));
                lsum[i] += term;
                pcnt[i] += 1;
            }
        }
    }

#pragma unroll
    for (int off = 1; off < 16; off <<= 1) {
#pragma unroll
        for (int i = 0; i < 8; ++i) {
            lsum[i] += __shfl_xor(lsum[i], off);
            pcnt[i] += __shfl_xor(pcnt[i], off);
        }
    }
    if (lmod == 0) {
#pragma unroll
        for (int i = 0; i < 8; ++i) {
            s_l[wave][8 * half + i] = lsum[i];
            s_c[wave][8 * half + i] = pcnt[i];
        }
    }
    __syncthreads();
    if (wave == 0 && lane < 16) {
        float ls = 0.0f; int pc = 0;
#pragma unroll
        for (int w = 0; w < NWAVES; ++w) { ls += s_l[w][lane]; pc += s_c[w][lane]; }
        float row_loss = (pc > 0) ? ls / (float)pc : 0.0f;
        // Sum the 16 row losses (lanes 0-15 all active: masks stay in-range).
#pragma unroll
        for (int off = 1; off < 16; off <<= 1)
            row_loss += __shfl_xor(row_loss, off);
        if (lane == 0)
            atomicAdd(out, row_loss / (float)Mtot);
    }
}

extern "C" void kernel_launch(void* const* d_in, const int* in_sizes, int n_in,
                              void* d_out, int out_size, void* d_ws, size_t ws_size,
                              hipStream_t stream) {
    const float* f     = (const float*)d_in[0];
    const int*   label = (const int*)d_in[1];
    float*       out   = (float*)d_out;

    const int Mtot = in_sizes[0] / D;       // 2N = 8192
    zero_out_kernel<<<1, 32, 0, stream>>>(out);
    scl_loss_kernel<<<Mtot / 16, NWAVES * 32, 0, stream>>>(f, label, out, Mtot);
}